// Conv_6133213298844
// MI455X (gfx1250) — compile-verified
//
#include <hip/hip_runtime.h>
#include <stdint.h>

// 3x3 SAME conv, 1x1x8192x8192 fp32, zero padding.
// Memory-bound (AI ~ 2.25 FLOP/B): design targets the 23.3 TB/s HBM roofline.
// CDNA5 paths used: global_load_async_to_lds_b128 (ASYNCcnt) for tile staging,
// s_wait_asynccnt, b128 stores with non-temporal hint.

#define IMG      8192
#define TILE_W   256
#define TILE_H   16
#define HALO_W   4                       // 4-float alignment halo each side
#define LDSW     (TILE_W + 2 * HALO_W)   // 264 floats per LDS row
#define LDSROWS  (TILE_H + 2)            // 18 rows (1-row halo top/bottom)
#define NCHUNK   (LDSW / 4)              // 66 b128 chunks per LDS row

typedef float v4f __attribute__((ext_vector_type(4)));

__global__ __launch_bounds__(256) void conv3x3_tile_async(
    const float* __restrict__ img,
    const float* __restrict__ kern,
    float* __restrict__ out)
{
    __shared__ float smem[LDSROWS * LDSW];   // 19,008 B

    const int tid = threadIdx.x;
    const int c0  = blockIdx.x * TILE_W;
    const int r0  = blockIdx.y * TILE_H;

    // ---- Stage input tile (+halo) into LDS via async b128 copies ----------
    // Chunk is fully in-bounds or fully out-of-bounds by construction:
    // TILE_W | IMG and HALO_W == 4 floats == one b128.
    const int total = LDSROWS * NCHUNK;      // 1188 chunks
    for (int idx = tid; idx < total; idx += 256) {
        const int row = idx / NCHUNK;
        const int ch  = idx - row * NCHUNK;
        const int gr  = r0 - 1 + row;              // global row (with halo)
        const int gc  = c0 - HALO_W + ch * 4;      // global col of chunk
        float* lp = &smem[row * LDSW + ch * 4];
        if ((unsigned)gr < IMG && gc >= 0 && gc + 3 < IMG) {
            const float*   gp       = img + (size_t)gr * IMG + gc;
            const unsigned lds_addr = (unsigned)(uintptr_t)lp;   // addr[31:0] = LDS offset
            const uint64_t gaddr    = (uint64_t)(uintptr_t)gp;
            asm volatile("global_load_async_to_lds_b128 %0, %1, off"
                         :: "v"(lds_addr), "v"(gaddr)
                         : "memory");
        } else {
            // zero padding region
            v4f z = {0.f, 0.f, 0.f, 0.f};
            *(v4f*)lp = z;
        }
    }
    asm volatile("s_wait_asynccnt 0" ::: "memory");
    __syncthreads();

    // ---- Weights (uniform -> scalar loads) --------------------------------
    float w[9];
#pragma unroll
    for (int i = 0; i < 9; ++i) w[i] = kern[i];

    // ---- Compute: each thread does a 4-row x 4-col register block ---------
    const int cx = (tid & 63) * 4;   // tile-local col of first output
    const int ry = (tid >> 6) * 4;   // tile-local row of first output

    // 6x6 LDS patch covers rows ry-1..ry+4 (LDS rows ry..ry+5) and
    // cols cx-1..cx+4 (LDS cols cx+3..cx+8). Each element loaded once.
    float r[6][6];
#pragma unroll
    for (int i = 0; i < 6; ++i)
#pragma unroll
        for (int j = 0; j < 6; ++j)
            r[i][j] = smem[(ry + i) * LDSW + (cx + 3 + j)];

#pragma unroll
    for (int rr = 0; rr < 4; ++rr) {
        float a[4];
#pragma unroll
        for (int cc = 0; cc < 4; ++cc) {
            float s = 0.f;
#pragma unroll
            for (int i = 0; i < 3; ++i)
#pragma unroll
                for (int j = 0; j < 3; ++j)
                    s = fmaf(w[i * 3 + j], r[rr + i][cc + j], s);
            a[cc] = s;
        }
        v4f acc = {a[0], a[1], a[2], a[3]};
        // write-once output: non-temporal b128 store
        __builtin_nontemporal_store(
            acc, (v4f*)(out + (size_t)(r0 + ry + rr) * IMG + (c0 + cx)));
    }
}

extern "C" void kernel_launch(void* const* d_in, const int* in_sizes, int n_in,
                              void* d_out, int out_size, void* d_ws, size_t ws_size,
                              hipStream_t stream) {
    (void)in_sizes; (void)n_in; (void)d_ws; (void)ws_size; (void)out_size;
    const float* img  = (const float*)d_in[0];   // (1,1,8192,8192) fp32
    const float* kern = (const float*)d_in[1];   // (3,3) fp32
    float*       out  = (float*)d_out;           // (1,1,8192,8192) fp32

    dim3 grid(IMG / TILE_W, IMG / TILE_H);       // 32 x 512
    dim3 block(256);                             // 8 wave32
    conv3x3_tile_async<<<grid, block, 0, stream>>>(img, kern, out);
}